// SVDMModel_14568529068432
// MI455X (gfx1250) — compile-verified
//
#include <hip/hip_runtime.h>
#include <math.h>

typedef __attribute__((ext_vector_type(2))) float v2f;
typedef __attribute__((ext_vector_type(8))) float v8f;

#define LSZ 256
#define NPTS (4 * LSZ * LSZ)      // 262144 rows through the MLP
#define NTILES (NPTS / 16)        // 16384 row-tiles of 16
#define WAVES_PER_BLOCK 4
#define CROP 246
#define NPIX (4 * CROP * CROP)    // 242064 cropped output pixels

__device__ __forceinline__ v8f wmma_f32_k4(v2f a, v2f b, v8f c) {
  // D = A(16x4,f32) * B(4x16,f32) + C(16x16,f32)
  return __builtin_amdgcn_wmma_f32_16x16x4_f32(
      /*neg_a=*/false, a, /*neg_b=*/false, b,
      /*c_mod=*/(short)0, c, /*reuse_a=*/false, /*reuse_b=*/false);
}

// ---------------------------------------------------------------------------
// Kernel 1: fused MLP  coords[N,2] -> 200 -> 50 -> {offset[2], spek[5], spak0[4]}
// One wave per 16-row tile; fp32 WMMA 16x16x4 throughout.
// All global loads are unconditional (clamped index) + cndmask-to-zero, so
// there is no exec-mask churn around the load stream.
// ---------------------------------------------------------------------------
__global__ __launch_bounds__(32 * WAVES_PER_BLOCK) void mlp_kernel(
    const float* __restrict__ coords,
    const float* __restrict__ w1, const float* __restrict__ b1,   // [2,200],[200]
    const float* __restrict__ w2, const float* __restrict__ b2,   // [200,50],[50]
    const float* __restrict__ wo, const float* __restrict__ bo,   // [50,2],[2]
    const float* __restrict__ wse, const float* __restrict__ bse, // [50,5],[5]
    const float* __restrict__ wsa, const float* __restrict__ bsa, // [50,4],[4]
    float* __restrict__ outOffset,   // [N,2]
    float* __restrict__ outSpak,     // [N,4]
    float* __restrict__ wsSpek)      // [N,5] workspace
{
  __shared__ float lds[WAVES_PER_BLOCK * 16 * 208];   // 52 KB

  const int wave = threadIdx.x >> 5;
  const int lane = threadIdx.x & 31;
  const int half = lane >> 4;     // 0: K=0,1 (A) / rows M+0..7 (C); 1: K=2,3 / M+8..15
  const int n16  = lane & 15;
  const int tile = blockIdx.x * WAVES_PER_BLOCK + wave;
  float* L = lds + wave * (16 * 208);

  // ---- A1: coords tile [16 x K4], K=2 real, K=2..3 zero-padded -------------
  const int row = tile * 16 + n16;             // lanes 0-15 carry M=0..15
  const float cx = coords[row * 2 + 0];        // unconditional, always in-bounds
  const float cy = coords[row * 2 + 1];
  v2f a1;
  a1.x = (half == 0) ? cx : 0.f;
  a1.y = (half == 0) ? cy : 0.f;

  // ---- Layer 1: f1[16,200] = relu(A1 @ w1 + b1), 13 col tiles --------------
#pragma unroll
  for (int j = 0; j < 13; ++j) {
    const int col  = j * 16 + n16;
    const int colc = min(col, 199);
    const bool cv  = (col < 200);
    const float w1a = w1[colc];                // rows K=0 / K=1 of w1
    const float w1b = w1[200 + colc];
    const bool sel = (half == 0) && cv;        // K=2,3 pad and col pad -> 0
    v2f b;
    b.x = sel ? w1a : 0.f;
    b.y = sel ? w1b : 0.f;
    v8f c = {};
    c = wmma_f32_k4(a1, b, c);
    const float bias = cv ? b1[colc] : 0.f;
#pragma unroll
    for (int v = 0; v < 8; ++v) {
      float val = c[v] + bias;
      val = val > 0.f ? val : 0.f;
      L[(v + 8 * half) * 208 + col] = val;     // C-format -> LDS (row-major)
    }
  }
  __syncthreads();

  // ---- Layer 2: f2[16,50] = relu(f1 @ w2 + b2); K=200 = 50 x K4 ------------
  v8f f2t[4];
#pragma unroll
  for (int j2 = 0; j2 < 4; ++j2) {
    const int col  = j2 * 16 + n16;
    const int colc = min(col, 49);
    const bool cv  = (col < 50);
    v8f acc = {};
    for (int kk = 0; kk < 50; ++kk) {
      const int k = kk * 4 + 2 * half;         // lanes0-15: K=k0,k0+1; 16-31: k0+2,k0+3
      const v2f a = *(const v2f*)(L + n16 * 208 + k);   // 8B-aligned DS b64 load
      const float w2a = w2[k * 50 + colc];
      const float w2b = w2[(k + 1) * 50 + colc];
      v2f b;
      b.x = cv ? w2a : 0.f;
      b.y = cv ? w2b : 0.f;
      acc = wmma_f32_k4(a, b, acc);
    }
    const float bias = cv ? b2[colc] : 0.f;
#pragma unroll
    for (int v = 0; v < 8; ++v) {
      float val = acc[v] + bias;
      f2t[j2][v] = val > 0.f ? val : 0.f;
    }
  }
  __syncthreads();

  // ---- spill f2 (C-format) to LDS, stride 64; cols 50..63 are zeros --------
#pragma unroll
  for (int j2 = 0; j2 < 4; ++j2) {
    const int col = j2 * 16 + n16;
#pragma unroll
    for (int v = 0; v < 8; ++v)
      L[(v + 8 * half) * 64 + col] = f2t[j2][v];
  }
  __syncthreads();

  // ---- Fused heads: combined B [52 x 16] = [wo | wse | wsa | pad] ----------
  const float* hptr; int hstride, hcol; bool hv;
  if (n16 < 2)       { hptr = wo;  hstride = 2; hcol = n16;     hv = true;  }
  else if (n16 < 7)  { hptr = wse; hstride = 5; hcol = n16 - 2; hv = true;  }
  else if (n16 < 11) { hptr = wsa; hstride = 4; hcol = n16 - 7; hv = true;  }
  else               { hptr = wo;  hstride = 2; hcol = 0;       hv = false; }

  v8f acc = {};
#pragma unroll
  for (int kk = 0; kk < 13; ++kk) {            // K=50 padded to 52
    const int k  = kk * 4 + 2 * half;
    const int kc = min(k, 48);                 // keep kc, kc+1 in [0,49]
    const v2f a = *(const v2f*)(L + n16 * 64 + k);
    const float ha = hptr[kc * hstride + hcol];        // unconditional loads
    const float hb = hptr[(kc + 1) * hstride + hcol];
    v2f b;
    b.x = (hv && k < 50)       ? ha : 0.f;
    b.y = (hv && (k + 1) < 50) ? hb : 0.f;
    acc = wmma_f32_k4(a, b, acc);
  }
  float hbias = 0.f;
  if (n16 < 2)       hbias = bo[n16];
  else if (n16 < 7)  hbias = bse[n16 - 2];
  else if (n16 < 11) hbias = bsa[n16 - 7];

#pragma unroll
  for (int v = 0; v < 8; ++v) {
    const float val = acc[v] + hbias;
    const int r = tile * 16 + v + 8 * half;
    if (n16 < 2)        outOffset[r * 2 + n16]      = val;
    else if (n16 < 7)   wsSpek[r * 5 + (n16 - 2)]   = val;
    else if (n16 < 11)  outSpak[r * 4 + (n16 - 7)]  = val;
  }
}

// ---------------------------------------------------------------------------
// Kernel 2: grid-sample msi (bilinear, zeros, align_corners=True) + pre_msi,
// o_offset and spek_c crops. One thread per cropped pixel.
// ---------------------------------------------------------------------------
__global__ __launch_bounds__(256) void sample_kernel(
    const float* __restrict__ msi,        // [4,4,256,256]
    const float* __restrict__ offsetIn,   // [N,2]
    const float* __restrict__ spekIn,     // [N,5]
    float* __restrict__ o_msi,            // [4,4,246,246]
    float* __restrict__ o_offset,         // [4,246,246,2]
    float* __restrict__ pre_msi,          // [4,246,246]
    float* __restrict__ spek_c)           // [4,246,246,5]
{
  const int tid = blockIdx.x * blockDim.x + threadIdx.x;
  if (tid >= NPIX) return;
  const int x = tid % CROP;
  const int y = (tid / CROP) % CROP;
  const int b = tid / (CROP * CROP);
  const int ix = x + 5, iy = y + 5;
  const int r = b * 65536 + iy * 256 + ix;

  const float off0 = offsetIn[r * 2 + 0];
  const float off1 = offsetIn[r * 2 + 1];
  // base grid (align_corners linspace) + offset/LS*2
  const float gx = (-1.f + (2.f / 255.f) * (float)ix) + off0 * (2.f / 256.f);
  const float gy = (-1.f + (2.f / 255.f) * (float)iy) + off1 * (2.f / 256.f);
  const float fx = (gx + 1.f) * 0.5f * 255.f;
  const float fy = (gy + 1.f) * 0.5f * 255.f;
  const float x0f = floorf(fx), y0f = floorf(fy);
  const float wx1 = fx - x0f, wy1 = fy - y0f;
  const int x0 = (int)x0f, y0 = (int)y0f;
  const int x1 = x0 + 1, y1 = y0 + 1;

  const bool vx0 = (x0 >= 0) & (x0 < 256), vx1 = (x1 >= 0) & (x1 < 256);
  const bool vy0 = (y0 >= 0) & (y0 < 256), vy1 = (y1 >= 0) & (y1 < 256);
  const float w00 = (vy0 && vx0) ? (1.f - wy1) * (1.f - wx1) : 0.f;
  const float w01 = (vy0 && vx1) ? (1.f - wy1) * wx1         : 0.f;
  const float w10 = (vy1 && vx0) ? wy1 * (1.f - wx1)         : 0.f;
  const float w11 = (vy1 && vx1) ? wy1 * wx1                 : 0.f;
  const int xc0 = min(max(x0, 0), 255), xc1 = min(max(x1, 0), 255);
  const int yc0 = min(max(y0, 0), 255), yc1 = min(max(y1, 0), 255);

  float s[5];
#pragma unroll
  for (int j = 0; j < 5; ++j) s[j] = spekIn[r * 5 + j];

  const float* mb = msi + b * 4 * 65536;
  float pm = s[4];                          // the "ones" channel
#pragma unroll
  for (int c = 0; c < 4; ++c) {
    const float* mc = mb + c * 65536;
    const float val = w00 * mc[yc0 * 256 + xc0] + w01 * mc[yc0 * 256 + xc1]
                    + w10 * mc[yc1 * 256 + xc0] + w11 * mc[yc1 * 256 + xc1];
    o_msi[((b * 4 + c) * CROP + y) * CROP + x] = val;
    pm += val * s[c];
  }
  const int oo = (b * CROP + y) * CROP + x;
  pre_msi[oo] = pm;
  o_offset[oo * 2 + 0] = off0;
  o_offset[oo * 2 + 1] = off1;
#pragma unroll
  for (int j = 0; j < 5; ++j) spek_c[oo * 5 + j] = s[j];
}

// ---------------------------------------------------------------------------
// Kernel 3: spatially-variant 15x15 Gaussian conv of pan, stride 4, pad (6,8).
// Normalization folded into num/den. Cropped region -> all taps in-bounds.
// ---------------------------------------------------------------------------
__global__ __launch_bounds__(256) void conv_kernel(
    const float* __restrict__ pan,        // [4,1024,1024]
    const float* __restrict__ spakIn,     // [N,4]
    float* __restrict__ pre_pan)          // [4,246,246]
{
  const int tid = blockIdx.x * blockDim.x + threadIdx.x;
  if (tid >= NPIX) return;
  const int x = tid % CROP;
  const int y = (tid / CROP) % CROP;
  const int b = tid / (CROP * CROP);
  const int ix = x + 5, iy = y + 5;
  const int r = b * 65536 + iy * 256 + ix;

  const float s0 = spakIn[r * 4 + 0];
  const float s2 = spakIn[r * 4 + 2];
  const float s3 = spakIn[r * 4 + 3];
  // m = [[s0,0],[s2,s3]], inv = m^T m
  const float i00 = s0 * s0 + s2 * s2;
  const float i01 = s2 * s3;
  const float i11 = s3 * s3;

  const float* pb = pan + b * 1024 * 1024 + (4 * iy - 6) * 1024 + (4 * ix - 6);
  float num = 0.f, den = 0.f;
#pragma unroll
  for (int p = 0; p < 15; ++p) {
    const float gp = (float)(p - 7);
    const float ty = i00 * gp * gp;
    const float txy = 2.f * i01 * gp;
    const float* prow = pb + p * 1024;
    for (int q = 0; q < 15; ++q) {
      const float gq = (float)(q - 7);
      const float t = ty + txy * gq + i11 * gq * gq;
      const float ker = __expf(-0.5f * t);
      den += ker;
      num = fmaf(ker, prow[q], num);
    }
  }
  pre_pan[(b * CROP + y) * CROP + x] = num / den;
}

// ---------------------------------------------------------------------------
extern "C" void kernel_launch(void* const* d_in, const int* in_sizes, int n_in,
                              void* d_out, int out_size, void* d_ws, size_t ws_size,
                              hipStream_t stream) {
  const float* msi    = (const float*)d_in[0];
  const float* pan    = (const float*)d_in[1];
  const float* coords = (const float*)d_in[2];
  const float* w1  = (const float*)d_in[3];
  const float* b1  = (const float*)d_in[4];
  const float* w2  = (const float*)d_in[5];
  const float* b2  = (const float*)d_in[6];
  const float* wo  = (const float*)d_in[7];
  const float* bo  = (const float*)d_in[8];
  const float* wse = (const float*)d_in[9];
  const float* bse = (const float*)d_in[10];
  const float* wsa = (const float*)d_in[11];
  const float* bsa = (const float*)d_in[12];

  float* out      = (float*)d_out;
  float* o_msi    = out;                 // [4,4,246,246]  = 968256
  float* o_offset = out + 968256;        // [4,246,246,2]  = 484128
  float* pre_msi  = out + 1452384;       // [4,1,246,246]  = 242064
  float* pre_pan  = out + 1694448;       // [4,1,246,246]  = 242064
  float* offsetO  = out + 1936512;       // [4,65536,2]    = 524288
  float* spek_c   = out + 2460800;       // [4,246,246,5]  = 1210320
  float* spak0    = out + 3671120;       // [4,65536,4]    = 1048576

  float* wsSpek = (float*)d_ws;          // [4,65536,5] full (uncropped) spek

  mlp_kernel<<<NTILES / WAVES_PER_BLOCK, 32 * WAVES_PER_BLOCK, 0, stream>>>(
      coords, w1, b1, w2, b2, wo, bo, wse, bse, wsa, bsa,
      offsetO, spak0, wsSpek);

  const int blocks = (NPIX + 255) / 256;
  sample_kernel<<<blocks, 256, 0, stream>>>(msi, offsetO, wsSpek,
                                            o_msi, o_offset, pre_msi, spek_c);
  conv_kernel<<<blocks, 256, 0, stream>>>(pan, spak0, pre_pan);
}